// ODERSSM_64201171140659
// MI455X (gfx1250) — compile-verified
//
#include <hip/hip_runtime.h>

#define DEVI __device__ __forceinline__

namespace ode_rssm {

constexpr int L_ = 256;
constexpr int B_ = 1024;
constexpr int K_ = 128;
constexpr int S_ = 64;
constexpr int OBS_ = 16;
constexpr int UIN_ = 16;
constexpr int OHID_ = 256;

typedef __attribute__((ext_vector_type(2))) float v2f;
typedef __attribute__((ext_vector_type(8))) float v8f;

DEVI v8f zero8() {
  v8f z = {0.f, 0.f, 0.f, 0.f, 0.f, 0.f, 0.f, 0.f};
  return z;
}

// D = A(16x4) * B(4x16) + C in fp32 on the matrix pipe (CDNA5 only).
DEVI v8f wmma4(v2f a, v2f b, v8f c) {
  return __builtin_amdgcn_wmma_f32_16x16x4_f32(false, a, false, b, (short)0, c,
                                               false, false);
}

DEVI float sigm(float x) { return 1.0f / (1.0f + __expf(-x)); }

// acc += A(16 x Kt) * W(rows n0..n0+15, Kt cols)^T  — caller pre-offsets Wn.
// A rows at As (row stride lda); weight rows at Wn (row stride ldw).
DEVI v8f gemm_acc(const float* As, int lda, const float* Wn, int ldw, int Kt,
                  v8f acc, int i16, int hi) {
  const float* ap = As + i16 * lda + 2 * hi;
  const float* bp = Wn + i16 * ldw + 2 * hi;
  for (int k = 0; k < Kt; k += 4) {
    v2f a; a.x = ap[k]; a.y = ap[k + 1];
    v2f b; b.x = bp[k]; b.y = bp[k + 1];
    acc = wmma4(a, b, acc);
  }
  return acc;
}

struct StepParams {
  const float *dw1, *db1, *dw2, *db2, *wmu, *bmu, *wls, *bls;  // post/dblock
  const float *zw1, *zb1, *zw2, *zb2;                          // pz preprocess
  const float *wih, *bih, *whh, *bhh;                          // gru
  const float *ow1, *ob1, *ow2, *ob2;                          // ode
};

// Two-layer ReLU MLP: in_dim=16 -> 128 -> 128. One 16-row tile per wave.
__global__ __launch_bounds__(128) void pre_kernel(
    const float* __restrict__ X, int ldx, const float* __restrict__ W1,
    const float* __restrict__ B1, const float* __restrict__ W2,
    const float* __restrict__ B2, float* __restrict__ Y, int nrows) {
  __shared__ float st[4][16 * 128];
  const int wv = threadIdx.x >> 5;
  const int lane = threadIdx.x & 31;
  const int i16 = lane & 15, hi = lane >> 4;
  const int m0 = (blockIdx.x * 4 + wv) * 16;
  if (m0 >= nrows) return;
  float* s = st[wv];
  const float* Xr = X + (size_t)m0 * ldx;
  for (int nt = 0; nt < 8; ++nt) {
    v8f acc = gemm_acc(Xr, ldx, W1 + nt * 16 * 16, 16, 16, zero8(), i16, hi);
    float b = B1[nt * 16 + i16];
#pragma unroll
    for (int r = 0; r < 8; ++r)
      s[(r + 8 * hi) * 128 + nt * 16 + i16] = fmaxf(acc[r] + b, 0.0f);
  }
  float* Yr = Y + (size_t)m0 * 128;
  for (int nt = 0; nt < 8; ++nt) {
    v8f acc = gemm_acc(s, 128, W2 + nt * 16 * 128, 128, 128, zero8(), i16, hi);
    float b = B2[nt * 16 + i16];
#pragma unroll
    for (int r = 0; r < 8; ++r)
      Yr[(r + 8 * hi) * 128 + nt * 16 + i16] = fmaxf(acc[r] + b, 0.0f);
  }
}

__global__ void init_kernel(float* __restrict__ rh, float* __restrict__ h0,
                            float* __restrict__ rh0) {
  int i = blockIdx.x * blockDim.x + threadIdx.x;
  if (i < B_ * K_) h0[i] = 0.0f;
  if (i < B_ * 2 * K_) {
    rh[i] = 0.0f;
    rh0[i] = 0.0f;
  }
}

// One full scan step, fused. 4 waves (128 threads) cooperate on one 16-row
// batch tile; every GEMM's output n-tiles are split across the 4 waves with
// __syncthreads() between producer/consumer phases. Grid = 64 blocks.
__global__ __launch_bounds__(128) void step_kernel(
    int t, const float* __restrict__ xe, const float* __restrict__ ue,
    const float* __restrict__ eps, const float* __restrict__ ext,
    float* __restrict__ rh, float* __restrict__ o_mu, float* __restrict__ o_ls,
    float* __restrict__ o_z, float* __restrict__ o_h, float* __restrict__ o_rh,
    StepParams P) {
  __shared__ float s_in[16 * 256];  // c=[xe,h]; x=[ue,ze]; ODE 'a' staging
  __shared__ float bufB[16 * 384];  // tt; then rh_old (16x256); then h_eval
  __shared__ float s_h[16 * 128];   // t1 staging; then ODE state h
  __shared__ float s_z[16 * 64];    // z staging

  const int tid = threadIdx.x;
  const int wv = tid >> 5;          // wave id 0..3
  const int lane = tid & 31;
  const int i16 = lane & 15;
  const int hi = lane >> 4;
  const int gm0 = blockIdx.x * 16;

  // ---- phase 0: c = [x_embed_t, h_prev] (cooperative load) ----------------
  {
    const float* xrow = xe + ((size_t)t * B_ + gm0) * K_;
    for (int idx = tid; idx < 16 * K_; idx += 128) {
      int r = idx >> 7, c = idx & (K_ - 1);
      s_in[r * 256 + c] = xrow[r * K_ + c];
      s_in[r * 256 + K_ + c] = rh[(gm0 + r) * 256 + c];  // h = rh[:, :128]
    }
  }
  __syncthreads();

  // ---- phase 1: tt = tanh(c@w1^T+b1)*sig(c@w2^T+b2), 6 tiles per wave -----
  for (int nt = wv; nt < 24; nt += 4) {
    v8f g1 = gemm_acc(s_in, 256, P.dw1 + nt * 16 * 256, 256, 256, zero8(), i16, hi);
    v8f g2 = gemm_acc(s_in, 256, P.dw2 + nt * 16 * 256, 256, 256, zero8(), i16, hi);
    float b1v = P.db1[nt * 16 + i16];
    float b2v = P.db2[nt * 16 + i16];
#pragma unroll
    for (int r = 0; r < 8; ++r)
      bufB[(r + 8 * hi) * 384 + nt * 16 + i16] =
          tanhf(g1[r] + b1v) * sigm(g2[r] + b2v);
  }
  __syncthreads();

  // ---- phase 2: mu/ls tile j=wv (K=384); z; plus ue -> s_in[:, :128] ------
  {
    v8f muacc = gemm_acc(bufB, 384, P.wmu + wv * 16 * 384, 384, 384, zero8(), i16, hi);
    v8f lsacc = gemm_acc(bufB, 384, P.wls + wv * 16 * 384, 384, 384, zero8(), i16, hi);
    float bmuv = P.bmu[wv * 16 + i16];
    float blsv = P.bls[wv * 16 + i16];
#pragma unroll
    for (int r = 0; r < 8; ++r) {
      int mrow = r + 8 * hi;
      size_t gidx = ((size_t)t * B_ + gm0 + mrow) * S_ + wv * 16 + i16;
      float mu = muacc[r] + bmuv;
      float ls = lsacc[r] + blsv;
      float zv = mu + __expf(ls) * eps[gidx];
      o_mu[gidx] = mu;
      o_ls[gidx] = ls;
      o_z[gidx] = zv;
      s_z[mrow * S_ + wv * 16 + i16] = zv;
    }
    const float* urow = ue + ((size_t)t * B_ + gm0) * K_;
    for (int idx = tid; idx < 16 * K_; idx += 128) {
      int r = idx >> 7, c = idx & (K_ - 1);
      s_in[r * 256 + c] = urow[r * K_ + c];
    }
  }
  __syncthreads();

  // ---- phase 3: t1 = relu(z@zw1^T+b) (2 tiles/wave); rh_old -> bufB -------
  for (int nt = wv; nt < 8; nt += 4) {
    v8f acc = gemm_acc(s_z, S_, P.zw1 + nt * 16 * S_, S_, S_, zero8(), i16, hi);
    float b = P.zb1[nt * 16 + i16];
#pragma unroll
    for (int r = 0; r < 8; ++r)
      s_h[(r + 8 * hi) * K_ + nt * 16 + i16] = fmaxf(acc[r] + b, 0.0f);
  }
  for (int idx = tid; idx < 16 * 256; idx += 128) {
    int r = idx >> 8, c = idx & 255;
    bufB[r * 256 + c] = rh[(gm0 + r) * 256 + c];
  }
  __syncthreads();

  // ---- phase 4: ze = relu(t1@zw2^T+b) -> s_in[:, 128:] --------------------
  for (int nt = wv; nt < 8; nt += 4) {
    v8f acc = gemm_acc(s_h, K_, P.zw2 + nt * 16 * K_, K_, K_, zero8(), i16, hi);
    float b = P.zb2[nt * 16 + i16];
#pragma unroll
    for (int r = 0; r < 8; ++r)
      s_in[(r + 8 * hi) * 256 + K_ + nt * 16 + i16] = fmaxf(acc[r] + b, 0.0f);
  }
  __syncthreads();

  // ---- phase 5: GRU, 4 n-tiles per wave -----------------------------------
  for (int nt = wv; nt < 16; nt += 4) {
    int n0 = nt * 16;
    v8f gir = gemm_acc(s_in, 256, P.wih + n0 * 256, 256, 256, zero8(), i16, hi);
    v8f ghr = gemm_acc(bufB, 256, P.whh + n0 * 256, 256, 256, zero8(), i16, hi);
    v8f giz = gemm_acc(s_in, 256, P.wih + (n0 + 256) * 256, 256, 256, zero8(), i16, hi);
    v8f ghz = gemm_acc(bufB, 256, P.whh + (n0 + 256) * 256, 256, 256, zero8(), i16, hi);
    v8f gig = gemm_acc(s_in, 256, P.wih + (n0 + 512) * 256, 256, 256, zero8(), i16, hi);
    v8f ghg = gemm_acc(bufB, 256, P.whh + (n0 + 512) * 256, 256, 256, zero8(), i16, hi);
    float bir = P.bih[n0 + i16], bhr = P.bhh[n0 + i16];
    float biz = P.bih[256 + n0 + i16], bhz = P.bhh[256 + n0 + i16];
    float big = P.bih[512 + n0 + i16], bhg = P.bhh[512 + n0 + i16];
#pragma unroll
    for (int r = 0; r < 8; ++r) {
      int mrow = r + 8 * hi;
      float rg = sigm(gir[r] + bir + ghr[r] + bhr);
      float zg = sigm(giz[r] + biz + ghz[r] + bhz);
      float ng = tanhf(gig[r] + big + rg * (ghg[r] + bhg));
      float old = bufB[mrow * 256 + n0 + i16];
      float rhn = (1.0f - zg) * ng + zg * old;
      if (n0 < K_) {
        s_h[mrow * K_ + n0 + i16] = rhn;  // ODE initial state
      } else {
        int gc = n0 + i16;
        rh[(gm0 + mrow) * 256 + gc] = rhn;
        o_rh[((size_t)(t + 1) * B_ + gm0 + mrow) * 256 + gc] = rhn;
      }
    }
  }
  __syncthreads();

  // ---- phase 6: ODE, 8 RK4 steps, per-row dt ------------------------------
  float stepm[8];
#pragma unroll
  for (int r = 0; r < 8; ++r) {
    int mrow = r + 8 * hi;
    stepm[r] = ext[((size_t)t * B_ + gm0 + mrow) * (UIN_ + 1) + UIN_] * 0.125f;
  }
  float* s_he = bufB;  // 16x128 h_eval staging (rh_old fully consumed)

  // a = tanh(hin@ow1^T+b1) -> s_in (4 tiles per wave)
  auto a_stage = [&](const float* hin) {
    for (int nt = wv; nt < 16; nt += 4) {
      v8f acc = gemm_acc(hin, K_, P.ow1 + nt * 16 * K_, K_, K_, zero8(), i16, hi);
      float b = P.ob1[nt * 16 + i16];
#pragma unroll
      for (int r = 0; r < 8; ++r)
        s_in[(r + 8 * hi) * OHID_ + nt * 16 + i16] = tanhf(acc[r] + b);
    }
  };
  // f = a@ow2^T+b2 for this wave's 2 column tiles
  auto f_eval = [&](v8f* kk) {
#pragma unroll
    for (int q = 0; q < 2; ++q) {
      int nt = 2 * wv + q;
      v8f acc = gemm_acc(s_in, OHID_, P.ow2 + nt * 16 * OHID_, OHID_, OHID_,
                         zero8(), i16, hi);
      float b = P.ob2[nt * 16 + i16];
#pragma unroll
      for (int r = 0; r < 8; ++r) acc[r] += b;
      kk[q] = acc;
    }
  };
  // h_eval = h + coef*step*k on this wave's columns
  auto he_write = [&](const v8f* kk, float coef) {
#pragma unroll
    for (int q = 0; q < 2; ++q) {
      int col0 = (2 * wv + q) * 16;
#pragma unroll
      for (int r = 0; r < 8; ++r) {
        int mrow = r + 8 * hi, col = col0 + i16;
        s_he[mrow * K_ + col] = s_h[mrow * K_ + col] + coef * stepm[r] * kk[q][r];
      }
    }
  };

  for (int it = 0; it < 8; ++it) {
    v8f kk[2], ksum[2];
    a_stage(s_h);
    __syncthreads();
    f_eval(kk);  // k1
    ksum[0] = kk[0];
    ksum[1] = kk[1];
    he_write(kk, 0.5f);
    __syncthreads();
    a_stage(s_he);
    __syncthreads();
    f_eval(kk);  // k2
#pragma unroll
    for (int q = 0; q < 2; ++q)
#pragma unroll
      for (int r = 0; r < 8; ++r) ksum[q][r] += 2.0f * kk[q][r];
    he_write(kk, 0.5f);
    __syncthreads();
    a_stage(s_he);
    __syncthreads();
    f_eval(kk);  // k3
#pragma unroll
    for (int q = 0; q < 2; ++q)
#pragma unroll
      for (int r = 0; r < 8; ++r) ksum[q][r] += 2.0f * kk[q][r];
    he_write(kk, 1.0f);
    __syncthreads();
    a_stage(s_he);
    __syncthreads();
    f_eval(kk);  // k4
#pragma unroll
    for (int q = 0; q < 2; ++q) {
      int col0 = (2 * wv + q) * 16;
#pragma unroll
      for (int r = 0; r < 8; ++r) {
        ksum[q][r] += kk[q][r];
        int mrow = r + 8 * hi, col = col0 + i16;
        s_h[mrow * K_ + col] += (stepm[r] * (1.0f / 6.0f)) * ksum[q][r];
      }
    }
    __syncthreads();
  }

  // ---- write h, h_seq[t+1], rh_seq[t+1, :128] -----------------------------
  for (int idx = tid; idx < 16 * K_; idx += 128) {
    int r = idx >> 7, c = idx & (K_ - 1);
    float hv = s_h[r * K_ + c];
    int m = gm0 + r;
    rh[m * 256 + c] = hv;
    o_h[((size_t)(t + 1) * B_ + m) * K_ + c] = hv;
    o_rh[((size_t)(t + 1) * B_ + m) * 256 + c] = hv;
  }
}

}  // namespace ode_rssm

extern "C" void kernel_launch(void* const* d_in, const int* in_sizes, int n_in,
                              void* d_out, int out_size, void* d_ws,
                              size_t ws_size, hipStream_t stream) {
  using namespace ode_rssm;
  (void)out_size;
  (void)ws_size;
  const size_t LB = (size_t)L_ * B_;

  int i_ext = -1, i_obs = -1, i_eps = -1;
  int pidx[28];
  int np = 0;
  for (int i = 0; i < n_in; ++i) {
    int sz = in_sizes[i];
    if (sz == (int)(LB * (UIN_ + 1)) && i_ext < 0) { i_ext = i; continue; }
    if (sz == (int)(LB * OBS_) && i_obs < 0) { i_obs = i; continue; }
    if (sz == (int)(LB * S_) && i_eps < 0) { i_eps = i; continue; }
    if (np < 28) pidx[np++] = i;
  }
  if (i_ext < 0 || i_obs < 0 || i_eps < 0 || np != 28) return;

  const float* P[28];
  for (int i = 0; i < 28; ++i) P[i] = (const float*)d_in[pidx[i]];

  // Param groups: either dict-insertion order (pu,px,pz,gru,post,ode) or
  // alphabetically sorted (gru,ode,post,pu,px,pz). Detect via first size.
  const float *pu[4], *px[4], *pz[4], *gru[4], *post[8], *ode[4];
  if (in_sizes[pidx[0]] == 196608) {  // gru.w_ih first -> sorted order
    for (int i = 0; i < 4; ++i) gru[i] = P[i];
    for (int i = 0; i < 4; ++i) ode[i] = P[4 + i];
    for (int i = 0; i < 8; ++i) post[i] = P[8 + i];
    for (int i = 0; i < 4; ++i) pu[i] = P[16 + i];
    for (int i = 0; i < 4; ++i) px[i] = P[20 + i];
    for (int i = 0; i < 4; ++i) pz[i] = P[24 + i];
  } else {  // pu.w1 (2048) first -> insertion order
    for (int i = 0; i < 4; ++i) pu[i] = P[i];
    for (int i = 0; i < 4; ++i) px[i] = P[4 + i];
    for (int i = 0; i < 4; ++i) pz[i] = P[8 + i];
    for (int i = 0; i < 4; ++i) gru[i] = P[12 + i];
    for (int i = 0; i < 8; ++i) post[i] = P[16 + i];
    for (int i = 0; i < 4; ++i) ode[i] = P[24 + i];
  }

  float* out = (float*)d_out;
  float* o_mu = out;
  float* o_ls = o_mu + LB * S_;
  float* o_z = o_ls + LB * S_;
  float* o_h = o_z + LB * S_;
  float* o_rh = o_h + (size_t)(L_ + 1) * B_ * K_;
  float* o_ue = o_rh + (size_t)(L_ + 1) * B_ * 2 * K_;

  float* w_xe = (float*)d_ws;    // x_embed: L*B*128
  float* w_rh = w_xe + LB * K_;  // rh carry: B*256

  const float* ext = (const float*)d_in[i_ext];
  const float* obs = (const float*)d_in[i_obs];
  const float* eps = (const float*)d_in[i_eps];

  int nrow = (int)LB;
  dim3 gpre(nrow / 64);
  pre_kernel<<<gpre, 128, 0, stream>>>(ext, UIN_ + 1, pu[0], pu[1], pu[2],
                                       pu[3], o_ue, nrow);
  pre_kernel<<<gpre, 128, 0, stream>>>(obs, OBS_, px[0], px[1], px[2], px[3],
                                       w_xe, nrow);
  init_kernel<<<(B_ * 2 * K_ + 255) / 256, 256, 0, stream>>>(w_rh, o_h, o_rh);

  StepParams SP;
  SP.dw1 = post[0]; SP.db1 = post[1]; SP.dw2 = post[2]; SP.db2 = post[3];
  SP.wmu = post[4]; SP.bmu = post[5]; SP.wls = post[6]; SP.bls = post[7];
  SP.zw1 = pz[0]; SP.zb1 = pz[1]; SP.zw2 = pz[2]; SP.zb2 = pz[3];
  SP.wih = gru[0]; SP.bih = gru[1]; SP.whh = gru[2]; SP.bhh = gru[3];
  SP.ow1 = ode[0]; SP.ob1 = ode[1]; SP.ow2 = ode[2]; SP.ob2 = ode[3];

  for (int t = 0; t < L_; ++t) {
    step_kernel<<<B_ / 16, 128, 0, stream>>>(t, w_xe, o_ue, eps, ext, w_rh,
                                             o_mu, o_ls, o_z, o_h, o_rh, SP);
  }
}